// LRU_30743375904839
// MI455X (gfx1250) — compile-verified
//
#include <hip/hip_runtime.h>
#include <math.h>

// LRU block for MI455X (gfx1250, wave32).
// k1: fused LayerNorm + complex input projection via V_WMMA_F32_16X16X4_F32
// scan: 3-pass chunked diagonal complex scan (in-place over u -> h)
// k3: complex output projection (real part) + exact GELU + D-skip, WMMA f32,
//     double-buffered LDS with GLOBAL_LOAD_ASYNC_TO_LDS_B128 A-slab staging.

#define L_SEQ   2048
#define BATCH   8
#define DM      1024
#define DH      1024
#define MROWS   (L_SEQ * BATCH)     // 16384 rows for both GEMMs
#define LN_EPSF 1e-5f

#define TILE_M  128
#define TILE_N  64
#define KSLAB   32
#define NSLAB   (DH / KSLAB)        // 32
#define LDSP    36                  // padded LDS row stride (floats), bank-conflict free

#define NCH     (BATCH * DH)        // 8192 scan channels
#define NCHUNK  16
#define CHLEN   (L_SEQ / NCHUNK)    // 128 (= 2^7, for repeated-squaring of lambda)

typedef float v2f __attribute__((ext_vector_type(2)));
typedef float v4f __attribute__((ext_vector_type(4)));
typedef float v8f __attribute__((ext_vector_type(8)));

// Async global->LDS 16B copy (GVS mode: 64-bit SGPR base + 32-bit VGPR byte offset).
// Tracked by ASYNCcnt; LDS dest address is the low 32 bits of the flat pointer.
__device__ __forceinline__ void async_cp16(unsigned int lds_addr, const float* base,
                                           int voff_bytes)
{
    asm volatile("global_load_async_to_lds_b128 %0, %1, %2"
                 :: "v"(lds_addr), "v"(voff_bytes), "s"(base)
                 : "memory");
}
__device__ __forceinline__ void wait_async0()
{
    asm volatile("s_wait_asynccnt 0x0" ::: "memory");
}

// ---------------------------------------------------------------------------
// Kernel 1: x = LN(inp);  u = gamma * (x @ B^T)   (complex, planar re/im out)
// ---------------------------------------------------------------------------
__global__ __launch_bounds__(256)
void lru_k1_ln_inproj(const float* __restrict__ inp,
                      const float* __restrict__ Bmat,   // complex64 [DH][DM] interleaved
                      const float* __restrict__ plog,   // [3][DH]: nu_log, th_log, gma_log
                      const float* __restrict__ ln_w,
                      const float* __restrict__ ln_b,
                      float* __restrict__ u_re,
                      float* __restrict__ u_im)
{
    __shared__ float As [TILE_M][LDSP];
    __shared__ float Bre[TILE_N][LDSP];
    __shared__ float Bim[TILE_N][LDSP];
    __shared__ float s_sum[TILE_M * 2];
    __shared__ float s_sq [TILE_M * 2];
    __shared__ float s_mu [TILE_M];
    __shared__ float s_rs [TILE_M];

    const int tid  = threadIdx.x;
    const int lane = tid & 31;
    const int w    = tid >> 5;                 // wave 0..7 -> M strip
    const int m0   = blockIdx.y * TILE_M;
    const int n0   = blockIdx.x * TILE_N;
    const int mi   = lane & 15;
    const int kh   = lane >> 4;

    // ---- LayerNorm statistics: 2 threads per row, 512 elems each ----
    {
        const int row  = tid >> 1;
        const int half = tid & 1;
        const float* p = inp + (size_t)(m0 + row) * DM + half * 512;
        float s = 0.f, q = 0.f;
        #pragma unroll 4
        for (int i = 0; i < 512; i += 4) {
            v4f v = *(const v4f*)(p + i);
            s += v.x + v.y + v.z + v.w;
            q += v.x * v.x + v.y * v.y + v.z * v.z + v.w * v.w;
        }
        s_sum[tid] = s; s_sq[tid] = q;
    }
    __syncthreads();
    if (tid < TILE_M) {
        float s  = s_sum[2 * tid] + s_sum[2 * tid + 1];
        float q  = s_sq [2 * tid] + s_sq [2 * tid + 1];
        float mu = s * (1.0f / DM);
        float vr = q * (1.0f / DM) - mu * mu;
        s_mu[tid] = mu;
        s_rs[tid] = rsqrtf(vr + LN_EPSF);
    }

    v8f acc_re[4], acc_im[4];
    #pragma unroll
    for (int j = 0; j < 4; ++j)
        for (int e = 0; e < 8; ++e) { acc_re[j][e] = 0.f; acc_im[j][e] = 0.f; }

    for (int k0 = 0; k0 < DM; k0 += KSLAB) {
        __syncthreads();
        // A slab: 128x32, LayerNorm applied on the fly
        #pragma unroll
        for (int i = 0; i < 4; ++i) {
            int idx = i * 1024 + tid * 4;
            int row = idx >> 5, col = idx & 31;
            v4f v  = *(const v4f*)(inp  + (size_t)(m0 + row) * DM + k0 + col);
            v4f wv = *(const v4f*)(ln_w + k0 + col);
            v4f bv = *(const v4f*)(ln_b + k0 + col);
            float mu = s_mu[row], rs = s_rs[row];
            v4f xn;
            xn.x = (v.x - mu) * rs * wv.x + bv.x;
            xn.y = (v.y - mu) * rs * wv.y + bv.y;
            xn.z = (v.z - mu) * rs * wv.z + bv.z;
            xn.w = (v.w - mu) * rs * wv.w + bv.w;
            *(v4f*)&As[row][col] = xn;
        }
        // B slab: 64 rows (h) x 32 cols (d), split re/im, k-contiguous in LDS
        #pragma unroll
        for (int i = 0; i < 4; ++i) {
            int idx = i * 1024 + tid * 4;
            int row = idx >> 6, f = idx & 63;
            v4f v = *(const v4f*)(Bmat + 2 * ((size_t)(n0 + row) * DM + k0) + f);
            int kc = f >> 1;
            v2f br = { v.x, v.z };
            v2f bi = { v.y, v.w };
            *(v2f*)&Bre[row][kc] = br;
            *(v2f*)&Bim[row][kc] = bi;
        }
        __syncthreads();

        #pragma unroll
        for (int kk = 0; kk < KSLAB; kk += 4) {
            v2f a = *(const v2f*)&As[16 * w + mi][kk + 2 * kh];
            #pragma unroll
            for (int j = 0; j < 4; ++j) {
                v2f br = *(const v2f*)&Bre[16 * j + mi][kk + 2 * kh];
                v2f bi = *(const v2f*)&Bim[16 * j + mi][kk + 2 * kh];
                acc_re[j] = __builtin_amdgcn_wmma_f32_16x16x4_f32(
                    false, a, false, br, (short)0, acc_re[j], false, false);
                acc_im[j] = __builtin_amdgcn_wmma_f32_16x16x4_f32(
                    false, a, false, bi, (short)0, acc_im[j], false, false);
            }
        }
    }

    // Epilogue: scale by gamma = exp(gma_log), write planar u
    #pragma unroll
    for (int j = 0; j < 4; ++j) {
        int col = n0 + 16 * j + mi;
        float g = expf(plog[2 * DH + col]);
        #pragma unroll
        for (int r = 0; r < 8; ++r) {
            int m = m0 + 16 * w + r + 8 * kh;
            size_t o = (size_t)m * DH + col;
            u_re[o] = g * acc_re[j][r];
            u_im[o] = g * acc_im[j][r];
        }
    }
}

// ---------------------------------------------------------------------------
// Diagonal complex scan  h_t = lam*h_{t-1} + u_t, chunked 3-pass, in place.
// ---------------------------------------------------------------------------
__device__ __forceinline__ void lam_of(const float* plog, int h, float& lre, float& lim)
{
    float nu = expf(plog[h]);
    float th = expf(plog[DH + h]);
    float r  = expf(-nu);
    float s, c;
    sincosf(th, &s, &c);
    lre = r * c; lim = r * s;
}

__global__ __launch_bounds__(256)
void lru_scan_pass1(float* __restrict__ u_re, float* __restrict__ u_im,
                    float* __restrict__ car_re, float* __restrict__ car_im,
                    const float* __restrict__ plog)
{
    int gid = blockIdx.x * 256 + threadIdx.x;     // channel-major: coalesced
    int ch  = gid & (NCH - 1);
    int c   = gid >> 13;
    float lre, lim; lam_of(plog, ch & (DH - 1), lre, lim);
    size_t base = (size_t)c * CHLEN * NCH + ch;
    float hre = 0.f, him = 0.f;
    #pragma unroll 4
    for (int t = 0; t < CHLEN; ++t) {
        size_t o = base + (size_t)t * NCH;
        float ur = u_re[o], ui = u_im[o];
        float nr = fmaf(lre, hre, fmaf(-lim, him, ur));
        float ni = fmaf(lre, him, fmaf( lim, hre, ui));
        hre = nr; him = ni;
        u_re[o] = hre; u_im[o] = him;
    }
    car_re[c * NCH + ch] = hre;
    car_im[c * NCH + ch] = him;
}

__global__ __launch_bounds__(256)
void lru_scan_pass2(float* __restrict__ car_re, float* __restrict__ car_im,
                    const float* __restrict__ plog)
{
    int ch = blockIdx.x * 256 + threadIdx.x;
    float lre, lim; lam_of(plog, ch & (DH - 1), lre, lim);
    float pr = lre, pi = lim;                     // lam^CHLEN, CHLEN = 2^7
    #pragma unroll
    for (int i = 0; i < 7; ++i) {
        float nr = pr * pr - pi * pi;
        float ni = 2.f * pr * pi;
        pr = nr; pi = ni;
    }
    float sre = car_re[ch], sim = car_im[ch];
    #pragma unroll
    for (int c = 1; c < NCHUNK; ++c) {
        int o = c * NCH + ch;
        float br = car_re[o], bi = car_im[o];
        float nr = fmaf(pr, sre, fmaf(-pi, sim, br));
        float ni = fmaf(pr, sim, fmaf( pi, sre, bi));
        sre = nr; sim = ni;
        car_re[o] = sre; car_im[o] = sim;
    }
}

__global__ __launch_bounds__(256)
void lru_scan_pass3(float* __restrict__ u_re, float* __restrict__ u_im,
                    const float* __restrict__ car_re, const float* __restrict__ car_im,
                    const float* __restrict__ plog)
{
    int gid = blockIdx.x * 256 + threadIdx.x;
    int ch  = gid & (NCH - 1);
    int c   = 1 + (gid >> 13);
    float lre, lim; lam_of(plog, ch & (DH - 1), lre, lim);
    float cr = car_re[(c - 1) * NCH + ch];
    float ci = car_im[(c - 1) * NCH + ch];
    float pr = lre * cr - lim * ci;               // lam^1 * carry
    float pi = lre * ci + lim * cr;
    size_t base = (size_t)c * CHLEN * NCH + ch;
    #pragma unroll 4
    for (int t = 0; t < CHLEN; ++t) {
        size_t o = base + (size_t)t * NCH;
        u_re[o] += pr; u_im[o] += pi;
        float nr = lre * pr - lim * pi;
        float ni = lre * pi + lim * pr;
        pr = nr; pi = ni;
    }
}

// ---------------------------------------------------------------------------
// Kernel 3: out = gelu(Re(h @ C^T)) + inp * D
// Double-buffered LDS; A slabs staged with async global->LDS (ASYNCcnt),
// C slabs register-prefetched (need de-interleave + negate).
// ---------------------------------------------------------------------------
__global__ __launch_bounds__(256)
void lru_k3_outproj(const float* __restrict__ h_re,
                    const float* __restrict__ h_im,
                    const float* __restrict__ Cmat,   // complex64 [DM][DH] interleaved
                    const float* __restrict__ inp,
                    const float* __restrict__ Dvec,
                    float* __restrict__ out)
{
    __shared__ float Are[2][TILE_M][LDSP];
    __shared__ float Aim[2][TILE_M][LDSP];
    __shared__ float Cre[2][TILE_N][LDSP];
    __shared__ float Cim[2][TILE_N][LDSP];   // stored negated

    const int tid  = threadIdx.x;
    const int lane = tid & 31;
    const int w    = tid >> 5;
    const int m0   = blockIdx.y * TILE_M;
    const int n0   = blockIdx.x * TILE_N;
    const int mi   = lane & 15;
    const int kh   = lane >> 4;

    // Per-thread A staging coordinates (fixed across slabs): 4 x 16B per plane.
    int rowA[4], colA[4], offA[4];               // offA: global byte offset at k0=0
    unsigned int ldsRe[2][4], ldsIm[2][4];
    #pragma unroll
    for (int i = 0; i < 4; ++i) {
        int idx = i * 1024 + tid * 4;
        rowA[i] = idx >> 5;
        colA[i] = idx & 31;
        offA[i] = ((m0 + rowA[i]) * DH + colA[i]) * 4;
        ldsRe[0][i] = (unsigned int)(size_t)&Are[0][rowA[i]][colA[i]];
        ldsRe[1][i] = (unsigned int)(size_t)&Are[1][rowA[i]][colA[i]];
        ldsIm[0][i] = (unsigned int)(size_t)&Aim[0][rowA[i]][colA[i]];
        ldsIm[1][i] = (unsigned int)(size_t)&Aim[1][rowA[i]][colA[i]];
    }
    const int rowC = (tid * 4) >> 6 & 63;        // C staging coords (i folded below)
    (void)rowC;

    v8f acc[4];
    #pragma unroll
    for (int j = 0; j < 4; ++j)
        for (int e = 0; e < 8; ++e) acc[j][e] = 0.f;

    v4f cbuf[4];
    int crow[4], ckc[4];
    #pragma unroll
    for (int i = 0; i < 4; ++i) {
        int idx = i * 1024 + tid * 4;
        crow[i] = idx >> 6;
        ckc[i]  = (idx & 63) >> 1;
    }

    // ---- prologue: stage slab 0 into buffer 0 ----
    #pragma unroll
    for (int i = 0; i < 4; ++i) {
        async_cp16(ldsRe[0][i], h_re, offA[i]);
        async_cp16(ldsIm[0][i], h_im, offA[i]);
    }
    #pragma unroll
    for (int i = 0; i < 4; ++i) {
        int idx = i * 1024 + tid * 4;
        cbuf[i] = *(const v4f*)(Cmat + 2 * ((size_t)(n0 + crow[i]) * DH) + (idx & 63));
    }
    #pragma unroll
    for (int i = 0; i < 4; ++i) {
        v2f cr = {  cbuf[i].x,  cbuf[i].z };
        v2f ci = { -cbuf[i].y, -cbuf[i].w };
        *(v2f*)&Cre[0][crow[i]][ckc[i]] = cr;
        *(v2f*)&Cim[0][crow[i]][ckc[i]] = ci;
    }
    wait_async0();
    __syncthreads();

    for (int s = 0; s < NSLAB; ++s) {
        const int cur = s & 1, nxt = cur ^ 1;
        const int k0n = (s + 1) * KSLAB;

        if (s + 1 < NSLAB) {
            // issue async A staging for next slab: overlaps with WMMA below
            #pragma unroll
            for (int i = 0; i < 4; ++i) {
                async_cp16(ldsRe[nxt][i], h_re, offA[i] + k0n * 4);
                async_cp16(ldsIm[nxt][i], h_im, offA[i] + k0n * 4);
            }
            // prefetch next C slab into registers
            #pragma unroll
            for (int i = 0; i < 4; ++i) {
                int idx = i * 1024 + tid * 4;
                cbuf[i] = *(const v4f*)(Cmat + 2 * ((size_t)(n0 + crow[i]) * DH + k0n)
                                        + (idx & 63));
            }
        }

        // ---- compute on current buffer ----
        #pragma unroll
        for (int kk = 0; kk < KSLAB; kk += 4) {
            v2f ar = *(const v2f*)&Are[cur][16 * w + mi][kk + 2 * kh];
            v2f ai = *(const v2f*)&Aim[cur][16 * w + mi][kk + 2 * kh];
            #pragma unroll
            for (int j = 0; j < 4; ++j) {
                v2f br = *(const v2f*)&Cre[cur][16 * j + mi][kk + 2 * kh];
                acc[j] = __builtin_amdgcn_wmma_f32_16x16x4_f32(
                    false, ar, false, br, (short)0, acc[j], false, false);
            }
            #pragma unroll
            for (int j = 0; j < 4; ++j) {
                v2f bi = *(const v2f*)&Cim[cur][16 * j + mi][kk + 2 * kh];
                acc[j] = __builtin_amdgcn_wmma_f32_16x16x4_f32(
                    false, ai, false, bi, (short)0, acc[j], false, false);
            }
        }

        if (s + 1 < NSLAB) {
            // commit next C slab to the other buffer, close async pipe, rotate
            #pragma unroll
            for (int i = 0; i < 4; ++i) {
                v2f cr = {  cbuf[i].x,  cbuf[i].z };
                v2f ci = { -cbuf[i].y, -cbuf[i].w };
                *(v2f*)&Cre[nxt][crow[i]][ckc[i]] = cr;
                *(v2f*)&Cim[nxt][crow[i]][ckc[i]] = ci;
            }
            wait_async0();
            __syncthreads();
        }
    }

    // Epilogue: exact GELU + skip through D
    #pragma unroll
    for (int j = 0; j < 4; ++j) {
        int col  = n0 + 16 * j + mi;
        float dv = Dvec[col];
        #pragma unroll
        for (int r = 0; r < 8; ++r) {
            int m = m0 + 16 * w + r + 8 * kh;
            size_t o = (size_t)m * DM + col;
            float x = acc[j][r];
            float y = 0.5f * x * (1.0f + erff(x * 0.70710678118654752f));
            out[o] = y + inp[o] * dv;
        }
    }
}

// ---------------------------------------------------------------------------
extern "C" void kernel_launch(void* const* d_in, const int* in_sizes, int n_in,
                              void* d_out, int out_size, void* d_ws, size_t ws_size,
                              hipStream_t stream)
{
    const float* inp  = (const float*)d_in[0];
    const float* Bmat = (const float*)d_in[1];   // complex64 -> interleaved float
    const float* Cmat = (const float*)d_in[2];
    const float* Dvec = (const float*)d_in[3];
    const float* plog = (const float*)d_in[4];
    const float* lnw  = (const float*)d_in[5];
    const float* lnb  = (const float*)d_in[6];
    float* out = (float*)d_out;

    float* u_re   = (float*)d_ws;                         // 64 MB
    float* u_im   = u_re + (size_t)MROWS * DH;            // 64 MB
    float* car_re = u_im + (size_t)MROWS * DH;            // 512 KB
    float* car_im = car_re + NCH * NCHUNK;                // 512 KB

    dim3 blk(256);
    dim3 g1(DH / TILE_N, MROWS / TILE_M);   // (16, 128)
    lru_k1_ln_inproj<<<g1, blk, 0, stream>>>(inp, Bmat, plog, lnw, lnb, u_re, u_im);

    lru_scan_pass1<<<(NCH * NCHUNK) / 256, blk, 0, stream>>>(u_re, u_im, car_re, car_im, plog);
    lru_scan_pass2<<<NCH / 256, blk, 0, stream>>>(car_re, car_im, plog);
    lru_scan_pass3<<<(NCH * (NCHUNK - 1)) / 256, blk, 0, stream>>>(u_re, u_im, car_re, car_im, plog);

    dim3 g3(DM / TILE_N, MROWS / TILE_M);   // (16, 128)
    lru_k3_outproj<<<g3, blk, 0, stream>>>(u_re, u_im, Cmat, inp, Dvec, out);
}